// OneHopGCNNormNodeLabelAggregator_8108898255209
// MI455X (gfx1250) — compile-verified
//
#include <hip/hip_runtime.h>
#include <hip/hip_bf16.h>

#define NFEAT 64
#define TILE  256   // edges staged per block

// ---------------------------------------------------------------- deg init --
__global__ void k_deg_init(float* __restrict__ deg, int n) {
    int i = blockIdx.x * blockDim.x + threadIdx.x;
    if (i < n) deg[i] = 1.0f;  // self-loop contributes 1 to every node's degree
}

// ------------------------------------------------------------- deg count ---
__global__ void k_deg_count(const int* __restrict__ row, float* __restrict__ deg, int E) {
    int e = blockIdx.x * blockDim.x + threadIdx.x;
    if (e < E) unsafeAtomicAdd(&deg[row[e]], 1.0f);   // hardware global_atomic_add_f32
}

// ------------------------------------------------------- dinv = rsqrt(deg) --
__global__ void k_dinv(const float* __restrict__ deg, float* __restrict__ dinv, int n) {
    int i = blockIdx.x * blockDim.x + threadIdx.x;
    if (i < n) {
        float d = deg[i];
        dinv[i] = (d > 0.0f) ? __frsqrt_rn(d) : 0.0f;
    }
}

// ----------------------- self-loop term, also initializes out (no memset) ---
// out[i,:] = x[i,:] * dinv[i]^2   (processed as float2 per thread)
__global__ void k_selfloop(const float* __restrict__ x, const float* __restrict__ dinv,
                           float* __restrict__ out, int n) {
    int t = blockIdx.x * blockDim.x + threadIdx.x;   // over n * 32 float2 elements
    if (t < n * (NFEAT / 2)) {
        int node = t >> 5;                            // 32 float2 per node
        float s = dinv[node];
        float w = s * s;
        float2 xv = ((const float2*)x)[t];
        ((float2*)out)[t] = make_float2(xv.x * w, xv.y * w);
    }
}

// ------------------------------------------------------------ edge scatter --
// Block = 256 threads (8 wave32). Phase A: async-DMA 256 edges' row/col into
// LDS, compute per-edge weight into LDS. Phase B: wave-per-edge, lane l owns
// features 2l..2l+1 -> one b64 gather + two f32 atomic adds.
__global__ __launch_bounds__(256) void k_scatter(const int* __restrict__ ei,   // [2*E]
                                                 const float* __restrict__ x,
                                                 const float* __restrict__ dinv,
                                                 float* __restrict__ out, int E) {
    __shared__ int   s_row[TILE];
    __shared__ int   s_col[TILE];
    __shared__ float s_w[TILE];

    const int t    = threadIdx.x;
    const int base = blockIdx.x * TILE;
    const int nval = min(TILE, E - base);             // edges in this tile (>=1)

    // ---- Phase A: stage row/col via gfx1250 async global->LDS DMA ----------
    {
        int e = base + min(t, nval - 1);              // clamp tail lanes in-bounds
        // Flat pointer to __shared__ carries the LDS byte offset in addr[31:0].
        unsigned lds_r = (unsigned)(size_t)&s_row[t];
        unsigned lds_c = (unsigned)(size_t)&s_col[t];
        const int* gr = ei + e;                       // row stream
        const int* gc = ei + E + e;                   // col stream
        asm volatile("global_load_async_to_lds_b32 %0, %1, off\n\t"
                     "global_load_async_to_lds_b32 %2, %3, off"
                     :: "v"(lds_r), "v"(gr), "v"(lds_c), "v"(gc)
                     : "memory");
        asm volatile("s_wait_asynccnt 0" ::: "memory");
    }
    {   // per-edge symmetric-norm weight
        int r = s_row[t];
        int c = s_col[t];
        s_w[t] = dinv[r] * dinv[c];
    }
    __syncthreads();

    // ---- Phase B: wave-per-edge gather/scatter -----------------------------
    const int wave = t >> 5;
    const int lane = t & 31;
    const float2* __restrict__ x2 = (const float2*)x;
    for (int k = 0; k < 32; ++k) {
        int s = (wave << 5) + k;
        if (s >= nval) break;
        int   r = s_row[s];                           // LDS broadcast reads
        int   c = s_col[s];
        float w = s_w[s];
        float2 xv = x2[r * (NFEAT / 2) + lane];       // coalesced 256B gather
        float* dst = out + c * NFEAT + lane * 2;
        unsafeAtomicAdd(dst,     w * xv.x);           // global_atomic_add_f32
        unsafeAtomicAdd(dst + 1, w * xv.y);
    }
}

extern "C" void kernel_launch(void* const* d_in, const int* in_sizes, int n_in,
                              void* d_out, int out_size, void* d_ws, size_t ws_size,
                              hipStream_t stream) {
    const float* x  = (const float*)d_in[0];
    const int*   ei = (const int*)d_in[1];            // [2, E] flattened
    float*       out = (float*)d_out;

    const int n = in_sizes[0] / NFEAT;
    const int E = in_sizes[1] / 2;

    float* deg  = (float*)d_ws;                       // n floats
    float* dinv = deg + n;                            // n floats

    k_deg_init <<<(n + 255) / 256,        256, 0, stream>>>(deg, n);
    k_deg_count<<<(E + 255) / 256,        256, 0, stream>>>(ei, deg, E);
    k_dinv     <<<(n + 255) / 256,        256, 0, stream>>>(deg, dinv, n);
    k_selfloop <<<(n * 32 + 255) / 256,   256, 0, stream>>>(x, dinv, out, n);
    k_scatter  <<<(E + TILE - 1) / TILE,  256, 0, stream>>>(ei, x, dinv, out, E);
}